// Model_7310034338114
// MI455X (gfx1250) — compile-verified
//
#include <hip/hip_runtime.h>
#include <float.h>

typedef __attribute__((ext_vector_type(2))) float v2f;
typedef __attribute__((ext_vector_type(8))) float v8f;

__device__ __forceinline__ v8f wmma_f32(v2f a, v2f b, v8f c) {
  return __builtin_amdgcn_wmma_f32_16x16x4_f32(false, a, false, b, (short)0, c,
                                               false, false);
}

// ---------------------------------------------------------------------------
// Weight prep: transpose W[K,N] -> Wt[Np,K] with zero-padded rows, pad bias.
//   dph > 0 (head mode): padded row n -> head h=n/16, d=n%16; src col h*dph+d,
//                        valid iff d < dph.
//   dph == 0 (linear):   src col n, valid iff n < N.
// ---------------------------------------------------------------------------
__global__ __launch_bounds__(256)
void prep_weights(const float* __restrict__ W, const float* __restrict__ bias,
                  float* __restrict__ Wt, float* __restrict__ bias_p,
                  int K, int N, int dph, int Np)
{
  const int idx = blockIdx.x * blockDim.x + threadIdx.x;
  if (idx >= Np * K) return;
  const int n = idx / K;
  const int k = idx - n * K;
  int src; bool valid;
  if (dph > 0) {
    const int d = n & 15, h = n >> 4;
    src = h * dph + d;
    valid = (d < dph);
  } else {
    src = n;
    valid = (n < N);
  }
  const int srcc = valid ? src : 0;
  Wt[(size_t)n * K + k] = valid ? W[(size_t)k * N + srcc] : 0.0f;
  if (k == 0) bias_p[n] = valid ? bias[srcc] : 0.0f;
}

// ---------------------------------------------------------------------------
// GEMM: C = scale * (A[M,K] @ Wt^T + bias_p), Wt pre-transposed & pre-padded.
// One wave per 64x16 tile. Software-pipelined: loads for step i+1 issue before
// the WMMAs of step i, so matrix ops wait on loads a full iteration old.
// ---------------------------------------------------------------------------
__global__ __launch_bounds__(256)
void gemm_tn16(const float* __restrict__ A, const float* __restrict__ Wt,
               const float* __restrict__ bias_p, float* __restrict__ C,
               int M, int K, int ldc, int Nc, int tiles_n, float scale,
               int total_waves)
{
  const int lane = threadIdx.x & 31;
  const int wave = blockIdx.x * (blockDim.x >> 5) + (threadIdx.x >> 5);
  if (wave >= total_waves) return;               // wave-uniform

  const int tm = wave / tiles_n;
  const int tn = wave % tiles_n;
  const int half = lane >> 4;
  const int l16 = lane & 15;
  const int m0 = tm << 6;
  const int col = (tn << 4) + l16;               // padded col space

  const float bval = bias_p[col];
  v8f acc0, acc1, acc2, acc3;
#pragma unroll
  for (int r = 0; r < 8; ++r) { acc0[r] = bval; acc1[r] = bval; acc2[r] = bval; acc3[r] = bval; }

  const int kbase = half * 2;
  const float* a0 = A + (size_t)(m0 + l16) * K + kbase;
  const float* a1 = a0 + (size_t)16 * K;
  const float* a2 = a0 + (size_t)32 * K;
  const float* a3 = a0 + (size_t)48 * K;
  const float* bt = Wt + (size_t)col * K + kbase;

  const int K4 = K & ~3;                          // K4 >= 64 for all our shapes
  // prologue: fragment set 0
  v2f bb  = *(const v2f*)bt;
  v2f fa0 = *(const v2f*)a0;
  v2f fa1 = *(const v2f*)a1;
  v2f fa2 = *(const v2f*)a2;
  v2f fa3 = *(const v2f*)a3;
#pragma unroll 4
  for (int k0 = 4; k0 < K4; k0 += 4) {
    a0 += 4; a1 += 4; a2 += 4; a3 += 4; bt += 4;
    v2f nb  = *(const v2f*)bt;                    // issue next loads first
    v2f nf0 = *(const v2f*)a0;
    v2f nf1 = *(const v2f*)a1;
    v2f nf2 = *(const v2f*)a2;
    v2f nf3 = *(const v2f*)a3;
    acc0 = wmma_f32(fa0, bb, acc0);               // consume previous loads
    acc1 = wmma_f32(fa1, bb, acc1);
    acc2 = wmma_f32(fa2, bb, acc2);
    acc3 = wmma_f32(fa3, bb, acc3);
    bb = nb; fa0 = nf0; fa1 = nf1; fa2 = nf2; fa3 = nf3;
  }
  acc0 = wmma_f32(fa0, bb, acc0);                 // epilogue of pipeline
  acc1 = wmma_f32(fa1, bb, acc1);
  acc2 = wmma_f32(fa2, bb, acc2);
  acc3 = wmma_f32(fa3, bb, acc3);

  if (K4 < K) {                                   // peeled tail (K % 4 != 0)
    const int ka0 = K4 + kbase, ka1 = ka0 + 1;
    const float km0 = (ka0 < K) ? 1.0f : 0.0f;    // zeroing B side suffices
    const float km1 = (ka1 < K) ? 1.0f : 0.0f;
    const int kc0 = (ka0 < K) ? ka0 : (K - 1);
    const int kc1 = (ka1 < K) ? ka1 : (K - 1);
    const float* wrow = Wt + (size_t)col * K;
    v2f tb;
    tb.x = wrow[kc0] * km0;
    tb.y = wrow[kc1] * km1;
    const float* ab = A + (size_t)(m0 + l16) * K;
    v2f t0, t1, t2, t3;
    t0.x = ab[kc0];                 t0.y = ab[kc1];
    t1.x = ab[(size_t)16*K + kc0];  t1.y = ab[(size_t)16*K + kc1];
    t2.x = ab[(size_t)32*K + kc0];  t2.y = ab[(size_t)32*K + kc1];
    t3.x = ab[(size_t)48*K + kc0];  t3.y = ab[(size_t)48*K + kc1];
    acc0 = wmma_f32(t0, tb, acc0);
    acc1 = wmma_f32(t1, tb, acc1);
    acc2 = wmma_f32(t2, tb, acc2);
    acc3 = wmma_f32(t3, tb, acc3);
  }

  if (col < Nc) {
    float* cb = C + (size_t)m0 * ldc + col;
#pragma unroll
    for (int r = 0; r < 8; ++r) {
      const int row = r + half * 8;
      cb[(size_t)(row      ) * ldc] = acc0[r] * scale;
      cb[(size_t)(row + 16 ) * ldc] = acc1[r] * scale;
      cb[(size_t)(row + 32 ) * ldc] = acc2[r] * scale;
      cb[(size_t)(row + 48 ) * ldc] = acc3[r] * scale;
    }
  }
}

// ---------------------------------------------------------------------------
// Flash attention over head-padded Q/K/V (stride Dp = 16*nheads; pad cols are
// exact zeros). DCH = ceil(dph/4). Softmax denominator is carried inside the
// O accumulator via a virtual ones-column of V at col dph (wmma row sums).
// K frags for the next tile and V frags for this tile are prefetched before
// the softmax VALU section so memory latency hides behind it.
// ---------------------------------------------------------------------------
template <int DCH>
__global__ __launch_bounds__(256)
void flash_attn_pad(const float* __restrict__ Q, const float* __restrict__ Kmat,
                    const float* __restrict__ V, const int* __restrict__ mask,
                    float* __restrict__ ctx, int S, int D, int dph, int nheads)
{
  __shared__ float plds[8][16][20];
  const int lane = threadIdx.x & 31;
  const int w = threadIdx.x >> 5;
  const int gwave = blockIdx.x * (blockDim.x >> 5) + w;

  const int qtiles = S >> 4;
  const int b = gwave / (nheads * qtiles);
  const int rem = gwave % (nheads * qtiles);
  const int h = rem / qtiles;
  const int q0 = (rem % qtiles) << 4;

  const int half = lane >> 4;
  const int l16 = lane & 15;
  const int Dp = nheads << 4;
  const size_t bstride = (size_t)S * Dp;
  const float* Qh = Q    + (size_t)b * bstride + (h << 4);
  const float* Kh = Kmat + (size_t)b * bstride + (h << 4);
  const float* Vh = V    + (size_t)b * bstride + (h << 4);
  const int*   mb = mask + b * S;

  const float vone = (l16 == dph) ? 1.0f : 0.0f;  // ones-column for row sums

  v8f o;
  float mrun[8];
#pragma unroll
  for (int r = 0; r < 8; ++r) { o[r] = 0.0f; mrun[r] = -FLT_MAX; }

  // Q fragments: invariant across key tiles -> load once.
  v2f aq[DCH];
  {
    const float* qrow = Qh + (size_t)(q0 + l16) * Dp + half * 2;
#pragma unroll
    for (int c = 0; c < DCH; ++c) aq[c] = *(const v2f*)(qrow + c * 4);
  }

  // prologue: K fragments for tile 0
  v2f kb[DCH];
  {
    const float* krow = Kh + (size_t)l16 * Dp + half * 2;
#pragma unroll
    for (int c = 0; c < DCH; ++c) kb[c] = *(const v2f*)(krow + c * 4);
  }

  for (int k0 = 0; k0 < S; k0 += 16) {
    // ---- S = Q @ K^T (uses prefetched K frags) ----
    v8f s;
#pragma unroll
    for (int r = 0; r < 8; ++r) s[r] = 0.0f;
#pragma unroll
    for (int c = 0; c < DCH; ++c) s = wmma_f32(aq[c], kb[c], s);

    // ---- prefetch next-tile K frags (clamped on last iter, branch-free) ----
    const int kn = (k0 + 16 < S) ? (k0 + 16) : 0;
    const float* nkrow = Kh + (size_t)(kn + l16) * Dp + half * 2;
#pragma unroll
    for (int c = 0; c < DCH; ++c) kb[c] = *(const v2f*)(nkrow + c * 4);

    // ---- prefetch this tile's V fragments ----
    float vx[4], vy[4];
#pragma unroll
    for (int kc = 0; kc < 4; ++kc) {
      const float* vrow = Vh + (size_t)(k0 + kc * 4 + half * 2) * Dp;
      vx[kc] = vrow[l16];
      vy[kc] = vrow[Dp + l16];
    }
    const bool masked = (mb[k0 + l16] == 0);

    // ---- online softmax (latency of the loads above hides under this) ----
#pragma unroll
    for (int r = 0; r < 8; ++r) {
      float sv = masked ? -FLT_MAX : s[r];
      float rmax = sv;
#pragma unroll
      for (int off = 1; off < 16; off <<= 1)
        rmax = fmaxf(rmax, __shfl_xor(rmax, off, 16));
      const float newm = fmaxf(mrun[r], rmax);
      const float p = __expf(sv - newm);
      const float alpha = __expf(mrun[r] - newm);
      mrun[r] = newm;
      o[r] *= alpha;                              // also rescales the sum col
      plds[w][r + half * 8][l16] = p;
    }

    // ---- O += P @ [V | 1]  (col dph of O accumulates the denominator) ----
#pragma unroll
    for (int kc = 0; kc < 4; ++kc) {
      const int kk = kc * 4 + half * 2;
      v2f pa, bv;
      pa.x = plds[w][l16][kk];
      pa.y = plds[w][l16][kk + 1];
      bv.x = vx[kc] + vone;                       // pad cols are 0; col dph = 1
      bv.y = vy[kc] + vone;
      o = wmma_f32(pa, bv, o);
    }
  }

  // ---- normalize (broadcast denominator from lane dph) + store ctx ----
  float* crow = ctx + (size_t)b * S * D + (size_t)h * dph + l16;
#pragma unroll
  for (int r = 0; r < 8; ++r) {
    const float l = __shfl(o[r], dph, 16);        // row sum for row r+8*half
    const float val = o[r] / l;
    if (l16 < dph) {
      const int m = q0 + r + half * 8;
      crow[(size_t)m * D] = val;
    }
  }
}

// ---------------------------------------------------------------------------
extern "C" void kernel_launch(void* const* d_in, const int* in_sizes, int n_in,
                              void* d_out, int out_size, void* d_ws, size_t ws_size,
                              hipStream_t stream)
{
  (void)in_sizes; (void)n_in; (void)out_size; (void)ws_size;

  const float* x     = (const float*)d_in[0];
  const float* y     = (const float*)d_in[1];
  const int*   mask0 = (const int*)d_in[2];
  const int*   mask1 = (const int*)d_in[3];
  const float* q0w = (const float*)d_in[4];  const float* q0b = (const float*)d_in[5];
  const float* k0w = (const float*)d_in[6];  const float* k0b = (const float*)d_in[7];
  const float* v0w = (const float*)d_in[8];  const float* v0b = (const float*)d_in[9];
  const float* o0w = (const float*)d_in[10]; const float* o0b = (const float*)d_in[11];
  const float* q1w = (const float*)d_in[12]; const float* q1b = (const float*)d_in[13];
  const float* k1w = (const float*)d_in[14]; const float* k1b = (const float*)d_in[15];
  const float* v1w = (const float*)d_in[16]; const float* v1b = (const float*)d_in[17];
  const float* o1w = (const float*)d_in[18]; const float* o1b = (const float*)d_in[19];

  const int BS = 4, SL = 1024, D0 = 192, D1 = 66;
  const int H0 = 16, DPH0 = 12, H1 = 11, DPH1 = 6;
  const int M = BS * SL;                          // 4096
  const int Dp0 = H0 * 16;                        // 256
  const int Dp1 = H1 * 16;                        // 176
  const int No0 = ((D0 + 15) / 16) * 16;          // 192
  const int No1 = ((D1 + 15) / 16) * 16;          // 80
  const size_t p0  = (size_t)M * Dp0;
  const size_t p1  = (size_t)M * Dp1;
  const size_t sz0 = (size_t)M * D0;
  const size_t sz1 = (size_t)M * D1;

  float* ws = (float*)d_ws;
  float* Q0p = ws;          float* K0p = Q0p + p0;   float* V0p = K0p + p0;
  float* Q1p = V0p + p0;    float* K1p = Q1p + p1;   float* V1p = K1p + p1;
  float* C0  = V1p + p1;    float* C1  = C0 + sz0;
  float* wp  = C1 + sz1;                          // weight-prep area
  float* Wq0 = wp;              wp += (size_t)Dp0 * D0;
  float* Wk0 = wp;              wp += (size_t)Dp0 * D0;
  float* Wv0 = wp;              wp += (size_t)Dp0 * D0;
  float* Wo0 = wp;              wp += (size_t)No0 * D0;
  float* Wq1 = wp;              wp += (size_t)Dp1 * D1;
  float* Wk1 = wp;              wp += (size_t)Dp1 * D1;
  float* Wv1 = wp;              wp += (size_t)Dp1 * D1;
  float* Wo1 = wp;              wp += (size_t)No1 * D1;
  float* Bq0 = wp;              wp += Dp0;
  float* Bk0 = wp;              wp += Dp0;
  float* Bv0 = wp;              wp += Dp0;
  float* Bo0 = wp;              wp += No0;
  float* Bq1 = wp;              wp += Dp1;
  float* Bk1 = wp;              wp += Dp1;
  float* Bv1 = wp;              wp += Dp1;
  float* Bo1 = wp;              wp += No1;
  float* out0 = (float*)d_out;
  float* out1 = out0 + sz0;

  auto prep = [&](const float* W, const float* bias, float* Wt, float* bp,
                  int K, int N, int dph, int Np) {
    const int total = Np * K;
    prep_weights<<<(total + 255) / 256, 256, 0, stream>>>(W, bias, Wt, bp, K, N, dph, Np);
  };
  prep(q0w, q0b, Wq0, Bq0, D0, D0, DPH0, Dp0);
  prep(k0w, k0b, Wk0, Bk0, D0, D0, DPH0, Dp0);
  prep(v0w, v0b, Wv0, Bv0, D0, D0, DPH0, Dp0);
  prep(o0w, o0b, Wo0, Bo0, D0, D0, 0,    No0);
  prep(q1w, q1b, Wq1, Bq1, D1, D1, DPH1, Dp1);
  prep(k1w, k1b, Wk1, Bk1, D1, D1, DPH1, Dp1);
  prep(v1w, v1b, Wv1, Bv1, D1, D1, DPH1, Dp1);
  prep(o1w, o1b, Wo1, Bo1, D1, D1, 0,    No1);

  auto gemm = [&](const float* A, const float* Wt, const float* bp, float* C,
                  int K, int ldc, int Nc, int Np, float scale) {
    const int tiles_n = Np >> 4;
    const int total = (M >> 6) * tiles_n;
    gemm_tn16<<<(total + 7) >> 3, 256, 0, stream>>>(
        A, Wt, bp, C, M, K, ldc, Nc, tiles_n, scale, total);
  };

  const float inv_s0 = 0.28867513459481287f;      // 1/sqrt(12)
  const float inv_s1 = 0.40824829046386307f;      // 1/sqrt(6)

  // projections into head-padded layouts (Q scaled by 1/sqrt(dph), incl. bias)
  gemm(x, Wq0, Bq0, Q0p, D0, Dp0, Dp0, Dp0, inv_s0);
  gemm(x, Wk0, Bk0, K0p, D0, Dp0, Dp0, Dp0, 1.0f);
  gemm(x, Wv0, Bv0, V0p, D0, Dp0, Dp0, Dp0, 1.0f);
  gemm(y, Wq1, Bq1, Q1p, D1, Dp1, Dp1, Dp1, inv_s1);
  gemm(y, Wk1, Bk1, K1p, D1, Dp1, Dp1, Dp1, 1.0f);
  gemm(y, Wv1, Bv1, V1p, D1, Dp1, Dp1, Dp1, 1.0f);

  // attention (waves = bs*heads*(S/16); both divisible by 8)
  flash_attn_pad<3><<<(BS * H0 * (SL >> 4)) / 8, 256, 0, stream>>>(
      Q0p, K0p, V0p, mask0, C0, SL, D0, DPH0, H0);
  flash_attn_pad<2><<<(BS * H1 * (SL >> 4)) / 8, 256, 0, stream>>>(
      Q1p, K1p, V1p, mask1, C1, SL, D1, DPH1, H1);

  // output projections straight into d_out (out0 | out1)
  gemm(C0, Wo0, Bo0, out0, D0, D0, D0, No0, 1.0f);
  gemm(C1, Wo1, Bo1, out1, D1, D1, D1, No1, 1.0f);
}